// EGNN_8864812499608
// MI455X (gfx1250) — compile-verified
//
#include <hip/hip_runtime.h>
#include <hip/hip_bf16.h>

#define NN   100000
#define EE   1600000
#define HH   128
#define LL   5
#define GG   128
#define OUTD 128
#define BN_EPS 1e-5f

typedef float v2f __attribute__((ext_vector_type(2)));
typedef float v8f __attribute__((ext_vector_type(8)));

#if defined(__has_builtin)
#if __has_builtin(__builtin_amdgcn_tensor_load_to_lds) && __has_builtin(__builtin_amdgcn_s_wait_tensorcnt)
#define HAVE_TDM 1
#endif
#endif

#ifdef HAVE_TDM
typedef unsigned int u32x4 __attribute__((ext_vector_type(4)));
typedef int          i32x8 __attribute__((ext_vector_type(8)));
typedef int          i32x4 __attribute__((ext_vector_type(4)));

// Issue one 2D TDM tile load: rows x rowlen f32 elements, source row stride
// stride_elems, into LDS at byte offset lds_off (tile stored contiguously,
// row-major). Tensor window == tile (always fully in bounds here).
// D# layout per CDNA5 ISA.
// This toolchain's builtin takes 6 args (clang-23 / therock-10.0 lane):
//   (u32x4 group0, i32x8 group1, i32x4 group2, i32x4 group3, i32x8 extra, i32 cpol)
__device__ __forceinline__ void tdm_load_2d(
    unsigned lds_off, const float* gsrc,
    unsigned rows, unsigned rowlen, unsigned stride_elems)
{
    unsigned long long ga = (unsigned long long)(uintptr_t)gsrc;
    u32x4 g0;
    g0[0] = 1u;                                   // count=1 (valid user D#)
    g0[1] = lds_off;                              // lds_addr (bytes)
    g0[2] = (unsigned)ga;                         // global_addr[31:0]
    g0[3] = (unsigned)((ga >> 32) & 0x1FFFFFFull) // global_addr[56:32]
          | (2u << 30);                           // type=2 ("image")
    i32x8 g1;
    g1[0] = (int)(2u << 16);                      // data_size=4B; wg_mask=0
    g1[1] = (int)((rowlen & 0xFFFFu) << 16);      // tensor_dim0[15:0] @ bits[63:48]
    g1[2] = (int)((rows   & 0xFFFFu) << 16);      // tensor_dim1[15:0] @ bits[95:80]
    g1[3] = (int)((rowlen & 0xFFFFu) << 16);      // tile_dim0 @ bits[127:112]
    g1[4] = (int)(rows & 0xFFFFu);                // tile_dim1 @ bits[143:128]; tile_dim2=0
    g1[5] = (int)stride_elems;                    // tensor_dim0_stride[31:0]
    g1[6] = 0;
    g1[7] = 0;
    i32x4 gz4 = {0, 0, 0, 0};
    i32x8 gz8 = {0, 0, 0, 0, 0, 0, 0, 0};
    __builtin_amdgcn_tensor_load_to_lds(g0, g1, gz4, gz4, gz8, 0);
}

__device__ __forceinline__ unsigned lds_off_of(const void* p) {
    // LDS flat addresses: low 32 bits are the LDS byte offset (ISA §10.2).
    return (unsigned)(uintptr_t)p;
}
#endif

// ---------------------------------------------------------------------------
// deg / norm precompute
// ---------------------------------------------------------------------------
__global__ void init_deg_kernel(float* __restrict__ deg) {
    int n = blockIdx.x * blockDim.x + threadIdx.x;
    if (n < NN) deg[n] = 1.0f;
}

__global__ void deg_count_kernel(const int* __restrict__ row, float* __restrict__ deg) {
    int e = blockIdx.x * blockDim.x + threadIdx.x;
    if (e < EE) atomicAdd(&deg[row[e]], 1.0f);
}

__global__ void dis_kernel(const float* __restrict__ deg, float* __restrict__ dis) {
    int n = blockIdx.x * blockDim.x + threadIdx.x;
    if (n < NN) dis[n] = rsqrtf(deg[n]);
}

__global__ void norm_kernel(const int* __restrict__ row, const int* __restrict__ col,
                            const float* __restrict__ dis, float* __restrict__ nrm) {
    int e = blockIdx.x * blockDim.x + threadIdx.x;
    if (e < EE) nrm[e] = dis[row[e]] * dis[col[e]];
}

__global__ void init_h_kernel(const int* __restrict__ x, const float* __restrict__ emb,
                              float* __restrict__ h) {
    long long idx = (long long)blockIdx.x * blockDim.x + threadIdx.x;
    if (idx >= (long long)NN * HH) return;
    int n = (int)(idx >> 7);
    int ch = (int)(idx & (HH - 1));
    h[idx] = emb[(size_t)x[n] * HH + ch];
}

__global__ void zero_kernel(float* __restrict__ p, long long n) {
    long long i = (long long)blockIdx.x * blockDim.x + threadIdx.x;
    if (i < n) p[i] = 0.0f;
}

// ---------------------------------------------------------------------------
// WMMA fp32 GEMM:  C[M x 128] = A[M x 128] @ B[128 x 128] + bias[128]
// Block = 256 threads = 8 waves; block owns a 16-row strip, wave w owns the
// 16x16 output tile at columns [16w,16w+16). K in 32-wide chunks, double-
// buffered in LDS. With TDM: wave 0 drives tensor_load_to_lds DMAs for the
// A (16x32) and B (32x128) tiles of the next chunk, s_wait_tensorcnt gates
// the workgroup barrier. Inner loop: V_WMMA_F32_16X16X4_F32.
// A-frag (16x4 f32): lane l -> row m0+(l&15), K = vgpr + 2*(l>>4)
// B-frag (4x16 f32): lane l -> col n0+(l&15), K = vgpr + 2*(l>>4)
// C/D   (16x16 f32): vgpr r -> row m0 + r + 8*(l>>4), col n0 + (l&15)
// ---------------------------------------------------------------------------
__global__ __launch_bounds__(256) void gemm_bias_kernel(
    const float* __restrict__ A, const float* __restrict__ B,
    const float* __restrict__ bias, float* __restrict__ C, int M)
{
    __shared__ float ldsA[2][16 * 32];
    __shared__ float ldsB[2][32 * 128];

    const int tid  = threadIdx.x;
    const int lane = tid & 31;
    const int wid  = tid >> 5;
    const int l16  = lane & 15;
    const int lh   = lane >> 4;
    const int m0   = blockIdx.x * 16;
    const int n0   = wid * 16;

    v8f c = {0.f, 0.f, 0.f, 0.f, 0.f, 0.f, 0.f, 0.f};

#ifdef HAVE_TDM
    if (wid == 0) {
        tdm_load_2d(lds_off_of(&ldsA[0][0]), A + (size_t)m0 * HH, 16, 32, HH);
        tdm_load_2d(lds_off_of(&ldsB[0][0]), B,                   32, 128, HH);
    }
    for (int kci = 0; kci < 4; ++kci) {
        const int kc = kci * 32;
        const int cb = kci & 1;
        if (wid == 0) __builtin_amdgcn_s_wait_tensorcnt(0);
        __syncthreads();                       // tile kci resident for all waves
        if (kci < 3 && wid == 0) {
            const int nb = cb ^ 1;             // other buffer: last read 1 iter ago, freed by barrier
            tdm_load_2d(lds_off_of(&ldsA[nb][0]),
                        A + (size_t)m0 * HH + (kc + 32), 16, 32, HH);
            tdm_load_2d(lds_off_of(&ldsB[nb][0]),
                        B + (size_t)(kc + 32) * HH,      32, 128, HH);
        }
        #pragma unroll
        for (int k = 0; k < 32; k += 4) {
            v2f a = *(const v2f*)(&ldsA[cb][l16 * 32 + k + 2 * lh]);
            v2f bf;
            bf.x = ldsB[cb][(k + 2 * lh)     * HH + n0 + l16];
            bf.y = ldsB[cb][(k + 2 * lh + 1) * HH + n0 + l16];
            c = __builtin_amdgcn_wmma_f32_16x16x4_f32(
                    false, a, false, bf, (short)0, c, false, false);
        }
    }
#else
    for (int kc = 0; kc < HH; kc += 32) {
        {
            int idx = tid * 2;
            int r  = idx >> 5;
            int cc = idx & 31;
            const float* src = A + (size_t)(m0 + r) * HH + kc + cc;
            ldsA[0][r * 32 + cc]     = src[0];
            ldsA[0][r * 32 + cc + 1] = src[1];
        }
        #pragma unroll
        for (int j = 0; j < 4; ++j) {
            int f   = tid + j * 256;
            int off = f * 4;
            int r   = off >> 7;
            int cc  = off & 127;
            float4 v = *(const float4*)(B + (size_t)(kc + r) * HH + cc);
            *(float4*)(&ldsB[0][off]) = v;
        }
        __syncthreads();
        #pragma unroll
        for (int k = 0; k < 32; k += 4) {
            v2f a = *(const v2f*)(&ldsA[0][l16 * 32 + k + 2 * lh]);
            v2f bf;
            bf.x = ldsB[0][(k + 2 * lh)     * HH + n0 + l16];
            bf.y = ldsB[0][(k + 2 * lh + 1) * HH + n0 + l16];
            c = __builtin_amdgcn_wmma_f32_16x16x4_f32(
                    false, a, false, bf, (short)0, c, false, false);
        }
        __syncthreads();
    }
#endif

    const int col = n0 + l16;
    const float bv = bias[col];
    #pragma unroll
    for (int r = 0; r < 8; ++r) {
        int rowi = m0 + r + 8 * lh;
        C[(size_t)rowi * HH + col] = c[r] + bv;
    }
}

// ---------------------------------------------------------------------------
// Edge message + scatter-add:
//   agg[col[e]] += norm[e] * relu(xl[row[e]] + edge_attr[e] @ We + be)
// 8 edges per block; thread = (edge, 4-channel group). We/be staged in LDS.
// ---------------------------------------------------------------------------
__global__ __launch_bounds__(256) void message_kernel(
    const int* __restrict__ row, const int* __restrict__ col,
    const float* __restrict__ edge_attr, const float* __restrict__ nrm,
    const float* __restrict__ xl, const float* __restrict__ We_i,
    const float* __restrict__ be_i, float* __restrict__ agg)
{
    __shared__ float sWe[7 * 128];
    __shared__ float sBe[128];

    const int tid = threadIdx.x;
    for (int idx = tid; idx < 1024; idx += 256) {
        if (idx < 896) sWe[idx] = We_i[idx];
        else           sBe[idx - 896] = be_i[idx - 896];
    }
    __syncthreads();

    const int eo  = tid >> 5;
    const int ch0 = (tid & 31) * 4;
    long long e = (long long)blockIdx.x * 8 + eo;
    if (e >= EE) return;

    const int   r    = row[e];
    const int   cdst = col[e];
    const float nv   = nrm[e];
    const float* ea  = edge_attr + e * 7;
    const float a0 = ea[0], a1 = ea[1], a2 = ea[2], a3 = ea[3];
    const float a4 = ea[4], a5 = ea[5], a6 = ea[6];

    float4 xv = *(const float4*)(xl + (size_t)r * HH + ch0);
    float* arow = agg + (size_t)cdst * HH + ch0;
    __builtin_prefetch(arow, 1, 0);   // global_prefetch_b8 on the scatter target

    #pragma unroll
    for (int j = 0; j < 4; ++j) {
        int ch = ch0 + j;
        float ec = sBe[ch]
                 + a0 * sWe[0 * 128 + ch] + a1 * sWe[1 * 128 + ch]
                 + a2 * sWe[2 * 128 + ch] + a3 * sWe[3 * 128 + ch]
                 + a4 * sWe[4 * 128 + ch] + a5 * sWe[5 * 128 + ch]
                 + a6 * sWe[6 * 128 + ch];
        float xj = (j == 0) ? xv.x : (j == 1) ? xv.y : (j == 2) ? xv.z : xv.w;
        float s = xj + ec;
        s = s > 0.f ? s : 0.f;
        atomicAdd(arow + j, nv * s);
    }
}

// ---------------------------------------------------------------------------
// Node update + BN statistics: pre = agg + relu(xl + root)/deg; accumulate
// per-channel sum / sumsq (128 channels = blockDim, grid-stride over nodes).
// ---------------------------------------------------------------------------
__global__ __launch_bounds__(128) void update_stats_kernel(
    const float* __restrict__ xl, const float* __restrict__ agg,
    const float* __restrict__ root_i, const float* __restrict__ deg,
    float* __restrict__ h, float* __restrict__ sums, float* __restrict__ sumsq)
{
    const int ch = threadIdx.x;
    const float rt = root_i[ch];
    float s = 0.f, s2 = 0.f;
    for (int n = blockIdx.x; n < NN; n += gridDim.x) {
        float xr = xl[(size_t)n * HH + ch] + rt;
        xr = xr > 0.f ? xr : 0.f;
        float pre = agg[(size_t)n * HH + ch] + xr / deg[n];
        h[(size_t)n * HH + ch] = pre;
        s  += pre;
        s2 += pre * pre;
    }
    atomicAdd(&sums[ch],  s);
    atomicAdd(&sumsq[ch], s2);
}

__global__ void bn_apply_kernel(float* __restrict__ h,
    const float* __restrict__ sums, const float* __restrict__ sumsq,
    const float* __restrict__ gamma_i, const float* __restrict__ beta_i,
    int do_relu)
{
    long long idx = (long long)blockIdx.x * blockDim.x + threadIdx.x;
    if (idx >= (long long)NN * HH) return;
    int ch = (int)(idx & (HH - 1));
    const float inv_n = 1.0f / (float)NN;
    float mean = sums[ch] * inv_n;
    float var  = sumsq[ch] * inv_n - mean * mean;
    float val = (h[idx] - mean) * rsqrtf(var + BN_EPS) * gamma_i[ch] + beta_i[ch];
    if (do_relu) val = val > 0.f ? val : 0.f;
    h[idx] = val;
}

// ---------------------------------------------------------------------------
// Global mean pool by graph
// ---------------------------------------------------------------------------
__global__ __launch_bounds__(128) void pool_kernel(
    const float* __restrict__ h, const int* __restrict__ batch,
    float* __restrict__ pool, float* __restrict__ cnt)
{
    int n  = blockIdx.x;
    int ch = threadIdx.x;
    int g  = batch[n];
    atomicAdd(&pool[(size_t)g * HH + ch], h[(size_t)n * HH + ch]);
    if (ch == 0) atomicAdd(&cnt[g], 1.0f);
}

__global__ void hg_kernel(const float* __restrict__ pool,
                          const float* __restrict__ cnt, float* __restrict__ hg)
{
    int idx = blockIdx.x * blockDim.x + threadIdx.x;
    if (idx >= GG * HH) return;
    int g = idx >> 7;
    float c = cnt[g];
    hg[idx] = pool[idx] / (c > 1.f ? c : 1.f);
}

// ---------------------------------------------------------------------------
extern "C" void kernel_launch(void* const* d_in, const int* in_sizes, int n_in,
                              void* d_out, int out_size, void* d_ws, size_t ws_size,
                              hipStream_t stream)
{
    const int*   x         = (const int*)d_in[0];
    const int*   ei        = (const int*)d_in[1];
    const int*   row       = ei;
    const int*   col       = ei + EE;
    const float* edge_attr = (const float*)d_in[2];
    const int*   batch     = (const int*)d_in[3];
    const float* node_emb  = (const float*)d_in[4];
    const float* W         = (const float*)d_in[5];
    const float* b         = (const float*)d_in[6];
    const float* We        = (const float*)d_in[7];
    const float* be        = (const float*)d_in[8];
    const float* root      = (const float*)d_in[9];
    const float* gamma     = (const float*)d_in[10];
    const float* beta      = (const float*)d_in[11];
    const float* Wout      = (const float*)d_in[12];
    const float* bout      = (const float*)d_in[13];
    float*       out       = (float*)d_out;

    char* ws = (char*)d_ws;
    size_t off = 0;
    auto carve = [&](size_t bytes) -> float* {
        float* p = (float*)(ws + off);
        off += (bytes + 255) & ~(size_t)255;
        return p;
    };
    float* h     = carve((size_t)NN * HH * 4);   // 51.2 MB
    float* xl    = carve((size_t)NN * HH * 4);   // 51.2 MB
    float* agg   = carve((size_t)NN * HH * 4);   // 51.2 MB
    float* deg   = carve((size_t)NN * 4);
    float* dis   = carve((size_t)NN * 4);
    float* nrm   = carve((size_t)EE * 4);        // 6.4 MB
    float* sums  = carve(HH * 4);
    float* sumsq = carve(HH * 4);
    float* pool  = carve((size_t)GG * HH * 4);
    float* cnt   = carve(GG * 4);
    float* hg    = carve((size_t)GG * HH * 4);

    const long long NH = (long long)NN * HH;
    const int nh_blocks = (int)((NH + 255) / 256);

    // degree / norm
    init_deg_kernel<<<(NN + 255) / 256, 256, 0, stream>>>(deg);
    deg_count_kernel<<<(EE + 255) / 256, 256, 0, stream>>>(row, deg);
    dis_kernel<<<(NN + 255) / 256, 256, 0, stream>>>(deg, dis);
    norm_kernel<<<(EE + 255) / 256, 256, 0, stream>>>(row, col, dis, nrm);

    // h = node_emb[x]
    init_h_kernel<<<nh_blocks, 256, 0, stream>>>(x, node_emb, h);

    for (int i = 0; i < LL; ++i) {
        // xl = h @ W[i] + b[i]   (WMMA fp32, TDM-staged tiles)
        gemm_bias_kernel<<<NN / 16, 256, 0, stream>>>(
            h, W + (size_t)i * HH * HH, b + (size_t)i * HH, xl, NN);

        zero_kernel<<<nh_blocks, 256, 0, stream>>>(agg, NH);
        zero_kernel<<<1, 256, 0, stream>>>(sums, HH);
        zero_kernel<<<1, 256, 0, stream>>>(sumsq, HH);

        message_kernel<<<EE / 8, 256, 0, stream>>>(
            row, col, edge_attr, nrm, xl,
            We + (size_t)i * 7 * HH, be + (size_t)i * HH, agg);

        update_stats_kernel<<<1024, 128, 0, stream>>>(
            xl, agg, root + (size_t)i * HH, deg, h, sums, sumsq);

        bn_apply_kernel<<<nh_blocks, 256, 0, stream>>>(
            h, sums, sumsq, gamma + (size_t)i * HH, beta + (size_t)i * HH,
            (i < LL - 1) ? 1 : 0);
    }

    // mean pool + output projection
    zero_kernel<<<(GG * HH + 255) / 256, 256, 0, stream>>>(pool, (long long)GG * HH);
    zero_kernel<<<1, 256, 0, stream>>>(cnt, GG);
    pool_kernel<<<NN, 128, 0, stream>>>(h, batch, pool, cnt);
    hg_kernel<<<(GG * HH + 255) / 256, 256, 0, stream>>>(pool, cnt, hg);

    gemm_bias_kernel<<<GG / 16, 256, 0, stream>>>(hg, Wout, bout, out, GG);
}